// QConv2D_31963146617133
// MI455X (gfx1250) — compile-verified
//
#include <hip/hip_runtime.h>
#include <math.h>

typedef __attribute__((ext_vector_type(2))) float v2f;
typedef __attribute__((ext_vector_type(8))) float v8f;

#define NQ 9
#define PATCHES 3136   /* 16 * 14 * 14 */
#define FILTERS 4
#define LAYERS 2
#define SIMS (PATCHES * FILTERS)
#define WAVES_PER_BLOCK 8
/* per (filter,layer) gate block: slotB 3*256 floats + laneA 3*1024 floats */
#define GATE_STRIDE 3840
#define SLOTB_OFF 0
#define LANEA_OFF 768

// forward map of the lane CNOT sub-chain CNOT(0,1)..CNOT(3,4) on row bits
__device__ __forceinline__ int lmap(int r) {
  if (r & 16) r ^= 8;
  if (r & 8)  r ^= 4;
  if (r & 4)  r ^= 2;
  if (r & 2)  r ^= 1;
  return r;
}
// forward map of the slot CNOT sub-chain CNOT(5,6)..CNOT(7,8) on column bits
__device__ __forceinline__ int smap(int c) {
  if (c & 8) c ^= 4;
  if (c & 4) c ^= 2;
  if (c & 2) c ^= 1;
  return c;
}

__device__ __forceinline__ v8f wmma4(v2f a, v2f b, v8f c) {
  return __builtin_amdgcn_wmma_f32_16x16x4_f32(false, a, false, b, (short)0, c,
                                               false, false);
}

/* ---------------- gate precompute: one block per (filter,layer) -------------
 * Builds, in WMMA fragment order:
 *  slotB: B = Gs^T  as {Re, Im, -Im}  (state is the A operand)
 *  laneA: A = Alam  as {Re, Im, -Im}  (state is the B operand)
 * where Gs = (U5(x)U6(x)U7(x)U8) * P_slotchain, Alam = (U0..U4 kron) * P_lanechain,
 * and Uq = RZ(w1)*RY(w0).
 */
__global__ __launch_bounds__(256) void build_gates(const float* __restrict__ wgt,
                                                   float* __restrict__ W) {
  const int f = blockIdx.x >> 1;
  const int l = blockIdx.x & 1;
  const int t = threadIdx.x;

  __shared__ float Ur[NQ][2][2], Ui[NQ][2][2];
  __shared__ float GsR[16][16], GsI[16][16];
  __shared__ float AR[32][32], AI[32][32];

  if (t < NQ) {
    float wy = wgt[(((f * LAYERS) + l) * NQ + t) * 2 + 0];
    float wz = wgt[(((f * LAYERS) + l) * NQ + t) * 2 + 1];
    float cy = cosf(0.5f * wy), sy = sinf(0.5f * wy);
    float er = cosf(0.5f * wz), ei = -sinf(0.5f * wz); /* e = exp(-i wz/2) */
    /* U = RZ*RY = [[e*cy, -e*sy],[conj(e)*sy, conj(e)*cy]] */
    Ur[t][0][0] =  er * cy;  Ui[t][0][0] =  ei * cy;
    Ur[t][0][1] = -er * sy;  Ui[t][0][1] = -ei * sy;
    Ur[t][1][0] =  er * sy;  Ui[t][1][0] = -ei * sy;
    Ur[t][1][1] =  er * cy;  Ui[t][1][1] = -ei * cy;
  }
  __syncthreads();

  { /* Gs[c][c'] = prod_q U_{5+q}[bit(c)][bit(smap(c'))] */
    int c = t >> 4, cp = t & 15, cc = smap(cp);
    float pr = 1.f, pi = 0.f;
#pragma unroll
    for (int qq = 0; qq < 4; ++qq) {
      int bi = (c  >> (3 - qq)) & 1;
      int bj = (cc >> (3 - qq)) & 1;
      float ur = Ur[5 + qq][bi][bj], ui = Ui[5 + qq][bi][bj];
      float nr = pr * ur - pi * ui, ni = pr * ui + pi * ur;
      pr = nr; pi = ni;
    }
    GsR[c][cp] = pr; GsI[c][cp] = pi;
  }
  /* Alam[r][r'] = prod_q U_q[bit(r)][bit(lmap(r'))] : 1024 entries, 4/thread */
  for (int e = 0; e < 4; ++e) {
    int idx = t + 256 * e;
    int r = idx >> 5, rp = idx & 31, rr = lmap(rp);
    float pr = 1.f, pi = 0.f;
#pragma unroll
    for (int q = 0; q < 5; ++q) {
      int bi = (r  >> (4 - q)) & 1;
      int bj = (rr >> (4 - q)) & 1;
      float ur = Ur[q][bi][bj], ui = Ui[q][bi][bj];
      float nr = pr * ur - pi * ui, ni = pr * ui + pi * ur;
      pr = nr; pi = ni;
    }
    AR[r][rp] = pr; AI[r][rp] = pi;
  }
  __syncthreads();

  float* Wb = W + (f * LAYERS + l) * GATE_STRIDE;

  /* slot B fragments: per chunk j (K0=4j), lane lam: {B[k][n], B[k+1][n]},
     k = 4j + 2*(lam>>4), n = lam&15, B = Gs^T  -> B[k][n] = Gs[n][k]   */
  for (int e = 0; e < 2; ++e) {
    int idx = t + 256 * e;
    if (idx < 384) {
      int m = idx / 128, rem = idx % 128, j = rem / 32, lam = rem % 32;
      int n = lam & 15, k = 4 * j + 2 * (lam >> 4);
      float v0, v1;
      if (m == 0)      { v0 =  GsR[n][k]; v1 =  GsR[n][k + 1]; }
      else if (m == 1) { v0 =  GsI[n][k]; v1 =  GsI[n][k + 1]; }
      else             { v0 = -GsI[n][k]; v1 = -GsI[n][k + 1]; }
      float* dst = Wb + SLOTB_OFF + m * 256 + (j * 32 + lam) * 2;
      dst[0] = v0; dst[1] = v1;
    }
  }
  /* lane A fragments: per (tile tt, chunk j), lane lam:
     {A[16tt + (lam&15)][k], A[..][k+1]}, k = 4j + 2*(lam>>4)           */
  for (int e = 0; e < 6; ++e) {
    int idx = t + 256 * e; /* 1536 float2 */
    int m = idx / 512, rem = idx % 512;
    int chunk = rem / 32, lam = rem % 32;
    int row = 16 * (chunk >> 3) + (lam & 15);
    int k = 4 * (chunk & 7) + 2 * (lam >> 4);
    float v0, v1;
    if (m == 0)      { v0 =  AR[row][k]; v1 =  AR[row][k + 1]; }
    else if (m == 1) { v0 =  AI[row][k]; v1 =  AI[row][k + 1]; }
    else             { v0 = -AI[row][k]; v1 = -AI[row][k + 1]; }
    float* dst = Wb + LANEA_OFF + m * 1024 + (chunk * 32 + lam) * 2;
    dst[0] = v0; dst[1] = v1;
  }
}

/* ---------------- main PQC kernel: one wave per (patch,filter) ------------- */
__global__ __launch_bounds__(256) void qconv_pqc(const float* __restrict__ x,
                                                 const float* __restrict__ W,
                                                 float* __restrict__ out) {
  __shared__ __align__(16) float Sre[WAVES_PER_BLOCK][512];
  __shared__ __align__(16) float Sim[WAVES_PER_BLOCK][512];

  const int tid  = threadIdx.x;
  const int wave = tid >> 5;
  const int lam  = tid & 31;
  const int lo   = lam & 15;
  const int hi   = lam >> 4;

  const int s  = blockIdx.x * WAVES_PER_BLOCK + wave;
  const int f  = s / PATCHES;
  const int p  = s % PATCHES;
  const int b  = p / 196;
  const int r2 = p % 196;
  const int ho = r2 / 14, wo = r2 % 14;

  /* ---- feature map: RX product state, built directly (no gates) ---- */
  float ch[9], sh[9];
#pragma unroll
  for (int i = 0; i < 3; ++i)
#pragma unroll
    for (int j = 0; j < 3; ++j) {
      float v = x[(b * 16 + ho + i) * 16 + (wo + j)];
      float h = 0.5f * v;
      ch[i * 3 + j] = cosf(h);
      sh[i * 3 + j] = sinf(h);
    }
  float lr = 1.f, li = 0.f; /* row factor over qubits 0..4, row = lam */
#pragma unroll
  for (int q = 0; q < 5; ++q) {
    int bit = (lam >> (4 - q)) & 1;
    float nr = bit ? (li * sh[q]) : (lr * ch[q]);
    float ni = bit ? (-lr * sh[q]) : (li * ch[q]);
    lr = nr; li = ni;
  }
  float* sre = Sre[wave];
  float* smm = Sim[wave];
#pragma unroll
  for (int c = 0; c < 16; ++c) {
    float pr = 1.f, pi = 0.f;
#pragma unroll
    for (int qq = 0; qq < 4; ++qq) {
      if ((c >> (3 - qq)) & 1) {
        float nr = pi * sh[5 + qq], ni = -pr * sh[5 + qq];
        pr = nr; pi = ni;
      } else { pr *= ch[5 + qq]; pi *= ch[5 + qq]; }
    }
    sre[lam * 16 + c] = lr * pr - li * pi;
    smm[lam * 16 + c] = lr * pi + li * pr;
  }
  __syncthreads();

  /* C45' predicate: column ^8 on rows r with bit0(lmap(r)) == 1 */
  const int g0 = lmap(lo) & 1;
  const int g1 = lmap(16 + lo) & 1;

  float zsum = 0.f;

#pragma unroll
  for (int l = 0; l < LAYERS; ++l) {
    const float* Wb = W + (f * LAYERS + l) * GATE_STRIDE;
    const v2f* slotB = (const v2f*)(Wb + SLOTB_OFF);
    const v2f* laneA = (const v2f*)(Wb + LANEA_OFF);

    /* ---- slot matmul: mid = (state w/ C45' column-xor) * Gs^T ---- */
    v8f aRe0 = {}, aIm0 = {}, aRe1 = {}, aIm1 = {};
#pragma unroll
    for (int j = 0; j < 4; ++j) {
      int kcol = 4 * j + 2 * hi;
      int c0 = g0 ? (kcol ^ 8) : kcol;
      int c1 = g1 ? (kcol ^ 8) : kcol;
      v2f mre0 = *(const v2f*)&sre[lo * 16 + c0];
      v2f mim0 = *(const v2f*)&smm[lo * 16 + c0];
      v2f mre1 = *(const v2f*)&sre[(16 + lo) * 16 + c1];
      v2f mim1 = *(const v2f*)&smm[(16 + lo) * 16 + c1];
      v2f bre = slotB[0 * 128 + j * 32 + lam];
      v2f bim = slotB[1 * 128 + j * 32 + lam];
      v2f bin = slotB[2 * 128 + j * 32 + lam];
      aRe0 = wmma4(mre0, bre, aRe0);  aRe0 = wmma4(mim0, bin, aRe0);
      aIm0 = wmma4(mre0, bim, aIm0);  aIm0 = wmma4(mim0, bre, aIm0);
      aRe1 = wmma4(mre1, bre, aRe1);  aRe1 = wmma4(mim1, bin, aRe1);
      aIm1 = wmma4(mre1, bim, aIm1);  aIm1 = wmma4(mim1, bre, aIm1);
    }
    __syncthreads();
    /* store mid (C/D layout -> canonical row-major) */
#pragma unroll
    for (int v = 0; v < 8; ++v) {
      int row0 = v + 8 * hi, row1 = 16 + v + 8 * hi;
      sre[row0 * 16 + lo] = aRe0[v];  smm[row0 * 16 + lo] = aIm0[v];
      sre[row1 * 16 + lo] = aRe1[v];  smm[row1 * 16 + lo] = aIm1[v];
    }
    __syncthreads();

    /* ---- lane matmul: new = Alam * mid ---- */
    v8f dRe0 = {}, dIm0 = {}, dRe1 = {}, dIm1 = {};
#pragma unroll
    for (int j = 0; j < 8; ++j) {
      int k = 4 * j + 2 * hi;
      v2f bre, bim;
      bre[0] = sre[k * 16 + lo];       bre[1] = sre[(k + 1) * 16 + lo];
      bim[0] = smm[k * 16 + lo];       bim[1] = smm[(k + 1) * 16 + lo];
      v2f ar0 = laneA[0 * 512 + (0 * 8 + j) * 32 + lam];
      v2f ai0 = laneA[1 * 512 + (0 * 8 + j) * 32 + lam];
      v2f an0 = laneA[2 * 512 + (0 * 8 + j) * 32 + lam];
      v2f ar1 = laneA[0 * 512 + (1 * 8 + j) * 32 + lam];
      v2f ai1 = laneA[1 * 512 + (1 * 8 + j) * 32 + lam];
      v2f an1 = laneA[2 * 512 + (1 * 8 + j) * 32 + lam];
      dRe0 = wmma4(ar0, bre, dRe0);  dRe0 = wmma4(an0, bim, dRe0);
      dIm0 = wmma4(ar0, bim, dIm0);  dIm0 = wmma4(ai0, bre, dIm0);
      dRe1 = wmma4(ar1, bre, dRe1);  dRe1 = wmma4(an1, bim, dRe1);
      dIm1 = wmma4(ar1, bim, dIm1);  dIm1 = wmma4(ai1, bre, dIm1);
    }

    if (l < LAYERS - 1) {
      __syncthreads();
#pragma unroll
      for (int v = 0; v < 8; ++v) {
        int row0 = v + 8 * hi, row1 = 16 + v + 8 * hi;
        sre[row0 * 16 + lo] = dRe0[v];  smm[row0 * 16 + lo] = dIm0[v];
        sre[row1 * 16 + lo] = dRe1[v];  smm[row1 * 16 + lo] = dIm1[v];
      }
      __syncthreads();
    } else {
      /* readout <Z_8>: this lane's amps all live in column lo */
      float acc = 0.f;
#pragma unroll
      for (int v = 0; v < 8; ++v)
        acc += dRe0[v] * dRe0[v] + dIm0[v] * dIm0[v] +
               dRe1[v] * dRe1[v] + dIm1[v] * dIm1[v];
      zsum = (lo & 1) ? -acc : acc;
    }
  }

  /* wave32 reduction over signed probabilities */
#pragma unroll
  for (int off = 16; off >= 1; off >>= 1)
    zsum += __shfl_xor(zsum, off, 32);
  if (lam == 0)
    out[((b * 14 + ho) * 14 + wo) * FILTERS + f] = tanhf(zsum);
}

extern "C" void kernel_launch(void* const* d_in, const int* in_sizes, int n_in,
                              void* d_out, int out_size, void* d_ws, size_t ws_size,
                              hipStream_t stream) {
  (void)in_sizes; (void)n_in; (void)out_size; (void)ws_size;
  const float* x   = (const float*)d_in[0];
  const float* wgt = (const float*)d_in[1];
  float* W = (float*)d_ws; /* needs 8*3840*4 = 120 KB */
  build_gates<<<FILTERS * LAYERS, 256, 0, stream>>>(wgt, W);
  qconv_pqc<<<SIMS / WAVES_PER_BLOCK, 256, 0, stream>>>(x, W, (float*)d_out);
}